// Compare_59141699666447
// MI455X (gfx1250) — compile-verified
//
#include <hip/hip_runtime.h>

// ---------------------------------------------------------------------------
// Graph U-Net (GraphConv + pool/unpool) for MI455X / gfx1250.
// Dense per-layer GEMMs use V_WMMA_F32_16X16X4_F32 (exact fp32 matrix op).
// Edge aggregation / pooling are division-free atomic scatter kernels
// (L2-resident working set: level-0 features are ~19MB << 192MB L2).
// ---------------------------------------------------------------------------

typedef float v2f __attribute__((ext_vector_type(2)));
typedef float v8f __attribute__((ext_vector_type(8)));

#define N0 150000
#define N1 37500
#define N2 9375
#define E0 2400000
#define E1 600000
#define E2 150000

// ---------------------------------------------------------------------------
// Feature preprocessing: x0 = [feat[:,0][tri6], feat[:,1][tri6], feat[:,2].flat9, geo]
// 22 real channels, padded to 24 (cols 22..23 = 0).
// blockDim = (24, 10): threadIdx.x = channel, threadIdx.y = node-in-block.
// ---------------------------------------------------------------------------
__global__ void build_x0_kernel(const float* __restrict__ feat,
                                const float* __restrict__ geo,
                                float* __restrict__ X0, int n)
{
    int c = threadIdx.x;                       // 0..23
    int i = blockIdx.x * blockDim.y + threadIdx.y;
    if (i >= n) return;
    // upper-tri indices of 3x3: (0,0)(0,1)(0,2)(1,1)(1,2)(2,2) -> 0,1,2,4,5,8
    const int tri6[6] = {0, 1, 2, 4, 5, 8};
    float v = 0.0f;
    if (c < 6)        v = feat[i * 27 + tri6[c]];
    else if (c < 12)  v = feat[i * 27 + 9 + tri6[c - 6]];
    else if (c < 21)  v = feat[i * 27 + 18 + (c - 12)];
    else if (c == 21) v = geo[i];
    X0[i * 24 + c] = v;
}

// ---------------------------------------------------------------------------
// Pack Ws/Wn/b into zero-padded (finP x foutP) layout so the WMMA loop is
// branch-free and padded rows/cols contribute exactly zero. (Tiny; 1D ok.)
// ---------------------------------------------------------------------------
__global__ void pack_w_kernel(const float* __restrict__ Ws,
                              const float* __restrict__ Wn,
                              const float* __restrict__ b,
                              float* __restrict__ WsP, float* __restrict__ WnP,
                              float* __restrict__ bP,
                              int fin, int fout, int finP, int foutP)
{
    int idx = blockIdx.x * blockDim.x + threadIdx.x;
    int per = finP * foutP;
    if (idx < per) {
        int k = idx / foutP, n = idx - k * foutP;
        WsP[idx] = (k < fin && n < fout) ? Ws[k * fout + n] : 0.0f;
    } else if (idx < 2 * per) {
        int t = idx - per;
        int k = t / foutP, n = t - k * foutP;
        WnP[t] = (k < fin && n < fout) ? Wn[k * fout + n] : 0.0f;
    } else if (idx < 2 * per + foutP) {
        int n = idx - 2 * per;
        bP[n] = (n < fout) ? b[n] : 0.0f;
    }
}

// ---------------------------------------------------------------------------
// Edge aggregation: agg[dst[e], c] += x[src[e], c]   (segment_sum)
// ei layout: ei[0:E]=src, ei[E:2E]=dst.
// blockDim.x == ch (division-free); threadIdx.y = edge within block.
// ---------------------------------------------------------------------------
__global__ void edge_agg_kernel(const float* __restrict__ X,
                                const int* __restrict__ ei,
                                float* __restrict__ AGG, int E, int ch)
{
    int c = threadIdx.x;
    int e = blockIdx.x * blockDim.y + threadIdx.y;
    if (e >= E) return;
    int s = ei[e];
    int d = ei[E + e];
    atomicAdd(&AGG[(size_t)d * ch + c], X[(size_t)s * ch + c]);
}

// ---------------------------------------------------------------------------
// Average pooling: scatter-sum + per-cluster count, then in-place divide.
// ---------------------------------------------------------------------------
__global__ void pool_scatter_kernel(const float* __restrict__ X,
                                    const int* __restrict__ cl,
                                    float* __restrict__ S, float* __restrict__ CNT,
                                    int Nf, int ch)
{
    int c = threadIdx.x;
    int i = blockIdx.x * blockDim.y + threadIdx.y;
    if (i >= Nf) return;
    int g = cl[i];
    atomicAdd(&S[(size_t)g * ch + c], X[(size_t)i * ch + c]);
    if (c == 0) atomicAdd(&CNT[g], 1.0f);
}

__global__ void pool_div_kernel(float* __restrict__ S, const float* __restrict__ CNT,
                                int Nc, int ch)
{
    int c = threadIdx.x;
    int i = blockIdx.x * blockDim.y + threadIdx.y;
    if (i >= Nc) return;
    size_t idx = (size_t)i * ch + c;
    S[idx] = S[idx] / fmaxf(CNT[i], 1.0f);
}

// ---------------------------------------------------------------------------
// Unpool + concat: O[i] = [ C[cluster[i]] (chC ch) | P[i] (chP ch) ]
// blockDim.x == chC + chP.
// ---------------------------------------------------------------------------
__global__ void concat_gather_kernel(const float* __restrict__ C,
                                     const int* __restrict__ cl,
                                     const float* __restrict__ P,
                                     float* __restrict__ O,
                                     int Nf, int chC, int chP)
{
    int ldo = chC + chP;
    int c = threadIdx.x;
    int i = blockIdx.x * blockDim.y + threadIdx.y;
    if (i >= Nf) return;
    O[(size_t)i * ldo + c] = (c < chC) ? C[(size_t)cl[i] * chC + c]
                                       : P[(size_t)i * chP + (c - chC)];
}

// ---------------------------------------------------------------------------
// GraphConv GEMM via V_WMMA_F32_16X16X4_F32:
//   OUT = act( X @ WsP + AGG @ WnP + b )
// Grid: blockIdx.x -> group of 8 M-tiles (one per wave), blockIdx.y -> N-tile.
// One wave computes a 64(M) x 16(N) tile: 4 accumulators; B fragment shared
// by 4 WMMAs. Full tiles take an unpredicated path with K unrolled by 2
// (8 WMMAs / iteration); only the boundary tile pays per-row predication.
// A-frag (fp32 16x4):  lane l -> A[row0+(l&15)][k+2*(l>>4) .. +1]  (one b64)
// B-frag (fp32 4x16):  lane l -> B[k+2*(l>>4)+{0,1}][col]
// C/D:                 VGPR j -> row j + 8*(l>>4), col l&15
// All branches around WMMA are wave-uniform -> EXEC all-ones as required.
// ---------------------------------------------------------------------------
__global__ __launch_bounds__(256)
void graphconv_wmma(const float* __restrict__ X, const float* __restrict__ AGG,
                    const float* __restrict__ WsP, const float* __restrict__ WnP,
                    const float* __restrict__ bP, float* __restrict__ OUT,
                    int Nrows, int finP, int foutP,
                    int ldout, int foutStore, int doRelu)
{
    const int lane = threadIdx.x & 31;
    const int wav  = threadIdx.x >> 5;
    const int tilesM = (Nrows + 63) >> 6;
    const int tm = blockIdx.x * 8 + wav;         // 8 waves / block
    if (tm >= tilesM) return;                    // wave-uniform
    const int r = lane & 15;
    const int h = lane >> 4;
    const int row0 = tm * 64;
    const int col  = (blockIdx.y << 4) + r;

    const float bv = bP[col];
    v8f acc[4];
#pragma unroll
    for (int m = 0; m < 4; ++m) {
#pragma unroll
        for (int j = 0; j < 8; ++j) acc[m][j] = bv;
    }

    const bool fullTile = (row0 + 64) <= Nrows;  // wave-uniform
    const v2f zero2 = {0.0f, 0.0f};

#pragma unroll 1
    for (int pass = 0; pass < 2; ++pass) {
        const float* __restrict__ Xp = pass ? AGG : X;
        const float* __restrict__ Wp = pass ? WnP : WsP;
        if (fullTile) {
            const float* __restrict__ arow0 = Xp + (size_t)(row0 + r) * finP;
            int k = 0;
            for (; k + 8 <= finP; k += 8) {
                const float* __restrict__ wrow = Wp + (size_t)(k + 2 * h) * foutP + col;
                v2f b0, b1;
                b0.x = wrow[0];
                b0.y = wrow[foutP];
                b1.x = wrow[4 * (size_t)foutP];
                b1.y = wrow[5 * (size_t)foutP];
                const float* __restrict__ ap = arow0 + k + 2 * h;
#pragma unroll
                for (int m = 0; m < 4; ++m) {
                    v2f a0 = *(const v2f*)(ap + (size_t)m * 16 * finP);
                    acc[m] = __builtin_amdgcn_wmma_f32_16x16x4_f32(
                        false, a0, false, b0, (short)0, acc[m], false, false);
                }
#pragma unroll
                for (int m = 0; m < 4; ++m) {
                    v2f a1 = *(const v2f*)(ap + (size_t)m * 16 * finP + 4);
                    acc[m] = __builtin_amdgcn_wmma_f32_16x16x4_f32(
                        false, a1, false, b1, (short)0, acc[m], false, false);
                }
            }
            for (; k < finP; k += 4) {           // tail (finP % 8 != 0)
                const int kk = k + 2 * h;
                v2f bf;
                bf.x = Wp[(size_t)(kk + 0) * foutP + col];
                bf.y = Wp[(size_t)(kk + 1) * foutP + col];
#pragma unroll
                for (int m = 0; m < 4; ++m) {
                    v2f af = *(const v2f*)(arow0 + (size_t)m * 16 * finP + kk);
                    acc[m] = __builtin_amdgcn_wmma_f32_16x16x4_f32(
                        false, af, false, bf, (short)0, acc[m], false, false);
                }
            }
        } else {
            for (int k = 0; k < finP; k += 4) {
                const int kk = k + 2 * h;
                v2f bf;
                bf.x = Wp[(size_t)(kk + 0) * foutP + col];
                bf.y = Wp[(size_t)(kk + 1) * foutP + col];
#pragma unroll
                for (int m = 0; m < 4; ++m) {
                    const int row = row0 + m * 16 + r;
                    v2f af = zero2;
                    if (row < Nrows)
                        af = *(const v2f*)(Xp + (size_t)row * finP + kk);
                    acc[m] = __builtin_amdgcn_wmma_f32_16x16x4_f32(
                        false, af, false, bf, (short)0, acc[m], false, false);
                }
            }
        }
    }

    const bool colOK = col < foutStore;
#pragma unroll
    for (int m = 0; m < 4; ++m) {
#pragma unroll
        for (int j = 0; j < 8; ++j) {
            const int row = row0 + m * 16 + h * 8 + j;
            if (row < Nrows && colOK) {
                float v = acc[m][j];
                if (doRelu) v = fmaxf(v, 0.0f);
                OUT[(size_t)row * ldout + col] = v;
            }
        }
    }
}

// ---------------------------------------------------------------------------
// Host orchestration
// ---------------------------------------------------------------------------
static inline int ceil4(int x)  { return (x + 3) & ~3; }
static inline int ceil16(int x) { return (x + 15) & ~15; }
static inline int cdiv(int a, int b) { return (a + b - 1) / b; }
static inline int rowsPerBlock(int ch) { int y = 256 / ch; return y < 1 ? 1 : y; }

static void conv_layer(const float* Xin, const int* ei, int E,
                       const float* Ws, const float* Wn, const float* b,
                       int fin, int fout, int N,
                       float* agg, float* wsP, float* wnP, float* bP,
                       float* out, int ldout, int foutStore, int relu,
                       hipStream_t st)
{
    const int finP  = ceil4(fin);
    const int foutP = ceil16(fout);
    hipMemsetAsync(agg, 0, (size_t)N * finP * sizeof(float), st);
    int tot = 2 * finP * foutP + foutP;
    pack_w_kernel<<<cdiv(tot, 256), 256, 0, st>>>(Ws, Wn, b, wsP, wnP, bP,
                                                  fin, fout, finP, foutP);
    int epb = rowsPerBlock(finP);
    edge_agg_kernel<<<cdiv(E, epb), dim3(finP, epb), 0, st>>>(Xin, ei, agg, E, finP);
    dim3 grid(cdiv(cdiv(N, 64), 8), foutP / 16);
    graphconv_wmma<<<grid, 256, 0, st>>>(Xin, agg, wsP, wnP, bP, out,
                                         N, finP, foutP, ldout, foutStore, relu);
}

static void pool_layer(const float* Xf, const int* cl, float* Sc, float* cnt,
                       int Nf, int Nc, int ch, hipStream_t st)
{
    hipMemsetAsync(Sc, 0, (size_t)Nc * ch * sizeof(float), st);
    hipMemsetAsync(cnt, 0, (size_t)Nc * sizeof(float), st);
    int y = rowsPerBlock(ch);
    pool_scatter_kernel<<<cdiv(Nf, y), dim3(ch, y), 0, st>>>(Xf, cl, Sc, cnt, Nf, ch);
    pool_div_kernel<<<cdiv(Nc, y), dim3(ch, y), 0, st>>>(Sc, cnt, Nc, ch);
}

extern "C" void kernel_launch(void* const* d_in, const int* in_sizes, int n_in,
                              void* d_out, int out_size, void* d_ws, size_t ws_size,
                              hipStream_t stream)
{
    const float* feat = (const float*)d_in[0];
    const float* geo  = (const float*)d_in[1];
    const int*   ei0  = (const int*)d_in[2];
    const int*   ei1  = (const int*)d_in[3];
    const int*   ei2  = (const int*)d_in[4];
    const int*   cl1  = (const int*)d_in[5];
    const int*   cl2  = (const int*)d_in[6];
    // params flattened in _DIMS order; each layer contributes Ws, Wn, b
    enum { L01, L02, L11, L12, L21, L22, L13, L14, L15, L16, L03, L04, L05, L06 };
    auto Wp = [&](int li, int j) -> const float* {
        return (const float*)d_in[7 + 3 * li + j];
    };

    // ---- workspace bump allocator (floats) ----
    float* base = (float*)d_ws;
    size_t off = 0;
    auto alloc = [&](size_t nflt) -> float* {
        float* p = base + off;
        off += (nflt + 63) & ~(size_t)63;
        return p;
    };
    float* x0   = alloc((size_t)N0 * 24);
    float* h0a  = alloc((size_t)N0 * 32);
    float* h0b  = alloc((size_t)N0 * 32);   // copy0 (persists)
    float* h0c  = alloc((size_t)N0 * 32);
    float* agg0 = alloc((size_t)N0 * 96);   // max finP at level 0
    float* c0   = alloc((size_t)N0 * 96);
    float* x1   = alloc((size_t)N1 * 32);
    float* h1a  = alloc((size_t)N1 * 64);
    float* h1b  = alloc((size_t)N1 * 64);   // copy1 (persists)
    float* h1c  = alloc((size_t)N1 * 64);
    float* agg1 = alloc((size_t)N1 * 192);
    float* c1   = alloc((size_t)N1 * 192);
    float* x2   = alloc((size_t)N2 * 64);
    float* h2a  = alloc((size_t)N2 * 128);
    float* h2b  = alloc((size_t)N2 * 128);
    float* agg2 = alloc((size_t)N2 * 128);
    float* cnt  = alloc((size_t)N1);        // pool counts (max of N1,N2)
    float* wsP  = alloc(16384);             // max finP*foutP = 128*128
    float* wnP  = alloc(16384);
    float* bP   = alloc(128);

    hipStream_t st = stream;

    // ---- preprocessing ----
    build_x0_kernel<<<cdiv(N0, 10), dim3(24, 10), 0, st>>>(feat, geo, x0, N0);

    // ---- encoder level 0 ----
    conv_layer(x0,  ei0, E0, Wp(L01,0), Wp(L01,1), Wp(L01,2), 22, 32, N0,
               agg0, wsP, wnP, bP, h0a, 32, 32, 1, st);
    conv_layer(h0a, ei0, E0, Wp(L02,0), Wp(L02,1), Wp(L02,2), 32, 32, N0,
               agg0, wsP, wnP, bP, h0b, 32, 32, 1, st);   // copy0

    // ---- pool 1 ----
    pool_layer(h0b, cl1, x1, cnt, N0, N1, 32, st);

    // ---- encoder level 1 ----
    conv_layer(x1,  ei1, E1, Wp(L11,0), Wp(L11,1), Wp(L11,2), 32, 64, N1,
               agg1, wsP, wnP, bP, h1a, 64, 64, 1, st);
    conv_layer(h1a, ei1, E1, Wp(L12,0), Wp(L12,1), Wp(L12,2), 64, 64, N1,
               agg1, wsP, wnP, bP, h1b, 64, 64, 1, st);   // copy1

    // ---- pool 2 ----
    pool_layer(h1b, cl2, x2, cnt, N1, N2, 64, st);

    // ---- bottleneck level 2 ----
    conv_layer(x2,  ei2, E2, Wp(L21,0), Wp(L21,1), Wp(L21,2), 64, 128, N2,
               agg2, wsP, wnP, bP, h2a, 128, 128, 1, st);
    conv_layer(h2a, ei2, E2, Wp(L22,0), Wp(L22,1), Wp(L22,2), 128, 128, N2,
               agg2, wsP, wnP, bP, h2b, 128, 128, 1, st);

    // ---- unpool 2 + concat -> 192 ch ----
    {
        int y = rowsPerBlock(192);
        concat_gather_kernel<<<cdiv(N1, y), dim3(192, y), 0, st>>>(h2b, cl2, h1b,
                                                                   c1, N1, 128, 64);
    }

    // ---- decoder level 1 ----
    conv_layer(c1,  ei1, E1, Wp(L13,0), Wp(L13,1), Wp(L13,2), 192, 64, N1,
               agg1, wsP, wnP, bP, h1a, 64, 64, 1, st);
    conv_layer(h1a, ei1, E1, Wp(L14,0), Wp(L14,1), Wp(L14,2), 64, 64, N1,
               agg1, wsP, wnP, bP, h1c, 64, 64, 1, st);
    conv_layer(h1c, ei1, E1, Wp(L15,0), Wp(L15,1), Wp(L15,2), 64, 64, N1,
               agg1, wsP, wnP, bP, h1a, 64, 64, 1, st);
    conv_layer(h1a, ei1, E1, Wp(L16,0), Wp(L16,1), Wp(L16,2), 64, 64, N1,
               agg1, wsP, wnP, bP, h1c, 64, 64, 1, st);

    // ---- unpool 1 + concat -> 96 ch ----
    {
        int y = rowsPerBlock(96);
        concat_gather_kernel<<<cdiv(N0, y), dim3(96, y), 0, st>>>(h1c, cl1, h0b,
                                                                  c0, N0, 64, 32);
    }

    // ---- decoder level 0 ----
    conv_layer(c0,  ei0, E0, Wp(L03,0), Wp(L03,1), Wp(L03,2), 96, 32, N0,
               agg0, wsP, wnP, bP, h0a, 32, 32, 1, st);
    conv_layer(h0a, ei0, E0, Wp(L04,0), Wp(L04,1), Wp(L04,2), 32, 32, N0,
               agg0, wsP, wnP, bP, h0c, 32, 32, 1, st);
    conv_layer(h0c, ei0, E0, Wp(L05,0), Wp(L05,1), Wp(L05,2), 32, 32, N0,
               agg0, wsP, wnP, bP, h0a, 32, 32, 1, st);
    // final layer: fout=3 (padded to 16 internally), store 3 cols, no relu
    conv_layer(h0a, ei0, E0, Wp(L06,0), Wp(L06,1), Wp(L06,2), 32, 3, N0,
               agg0, wsP, wnP, bP, (float*)d_out, 3, 3, 0, st);

    (void)in_sizes; (void)n_in; (void)out_size; (void)ws_size;
}